// MTGODE_86775519249022
// MI455X (gfx1250) — compile-verified
//
#include <hip/hip_runtime.h>
#include <hip/hip_bf16.h>

// ---------------- problem constants ----------------
#define GB   16          // batch
#define GL   12          // seq len
#define GN   2000        // graph nodes
#define GC   32          // hidden channels
#define GF   (GB*GC*GL)  // 6144  state columns, f = b*384 + c*12 + l
#define GSITES (GN*GB*GL) // 384000 per-(n,b,l) sites
#define GE   128
#define GCO  12

typedef __attribute__((ext_vector_type(2))) float v2f;
typedef __attribute__((ext_vector_type(8))) float v8f;

// ---------------- adjacency prep ----------------
__global__ __launch_bounds__(256) void rowsum_kernel(const float* __restrict__ adj,
                                                     float* __restrict__ rowsum) {
    __shared__ float sh[256];
    const int n = blockIdx.x;
    float s = 0.f;
    const float* p = adj + (size_t)n * GN;
    for (int m = threadIdx.x; m < GN; m += 256) s += p[m];
    sh[threadIdx.x] = s;
    __syncthreads();
    for (int o = 128; o > 0; o >>= 1) {
        if (threadIdx.x < o) sh[threadIdx.x] += sh[threadIdx.x + o];
        __syncthreads();
    }
    if (threadIdx.x == 0) rowsum[n] = sh[0];
}

__global__ __launch_bounds__(256) void colsum_kernel(const float* __restrict__ adj,
                                                     float* __restrict__ colsum) {
    const int n = blockIdx.x * 256 + threadIdx.x;
    if (n >= GN) return;
    float s = 0.f;
    for (int m = 0; m < GN; ++m) s += adj[(size_t)m * GN + n];   // lanes: consecutive n -> coalesced
    colsum[n] = s;
}

// A1[n,m] = (adj[n,m]+I)/(rowsum[n]+1) ; A2[m,n] = (adj[n,m]+I)/(colsum[m]+1)
__global__ __launch_bounds__(256) void build_adj_kernel(const float* __restrict__ adj,
                                                        const float* __restrict__ rowsum,
                                                        const float* __restrict__ colsum,
                                                        float* __restrict__ A1,
                                                        float* __restrict__ A2) {
    const size_t i = (size_t)blockIdx.x * 256 + threadIdx.x;
    if (i >= (size_t)GN * GN) return;
    const int n = (int)(i / GN);
    const int m = (int)(i % GN);
    const float a = adj[i] + (n == m ? 1.f : 0.f);
    A1[i] = a / (rowsum[n] + 1.f);
    A2[(size_t)m * GN + n] = a / (colsum[m] + 1.f);
}

// ---------------- start conv: x[B,L,N,2] -> H[n, f] ----------------
__global__ __launch_bounds__(256) void conv_start_kernel(const float* __restrict__ x,
                                                         const float* __restrict__ w,  // [32,2]
                                                         const float* __restrict__ bias,
                                                         float* __restrict__ H) {
    __shared__ float sw[64], sb[32];
    if (threadIdx.x < 64) sw[threadIdx.x] = w[threadIdx.x];
    if (threadIdx.x < 32) sb[threadIdx.x] = bias[threadIdx.x];
    __syncthreads();
    const int sid = blockIdx.x * 256 + threadIdx.x;
    if (sid >= GSITES) return;
    const int n = sid / (GB * GL);
    const int r = sid % (GB * GL);
    const int b = r / GL;
    const int l = r % GL;
    const float x0 = x[((size_t)((b * GL + l) * GN + n)) * 2 + 0];
    const float x1 = x[((size_t)((b * GL + l) * GN + n)) * 2 + 1];
    float* out = H + (size_t)n * GF + b * (GC * GL) + l;
#pragma unroll
    for (int c = 0; c < GC; ++c)
        out[c * GL] = sw[c * 2] * x0 + sw[c * 2 + 1] * x1 + sb[c];
}

// ---------------- WMMA Euler propagation step ----------------
// Y[n,f] = 0.75*X[n,f] + 0.25 * (A @ X)[n,f]     A: [GN,GN] row-major, X/Y: [GN,GF]
// Register tiling: 32(M) x 128(N) per wave -> 16 wmma per K-step sharing 8 B-frags.
#define NT 8
__global__ __launch_bounds__(256) void prop_step_kernel(const float* __restrict__ A,
                                                        const float* __restrict__ X,
                                                        float* __restrict__ Y) {
    const int lane  = threadIdx.x & 31;
    const int wave  = threadIdx.x >> 5;
    const int mPair = blockIdx.x;                 // 0..62  (M in pairs of 16-row subtiles)
    const int strip = blockIdx.y * 8 + wave;      // 0..47
    const int row16 = lane & 15;
    const int half  = lane >> 4;
    const int nBase = strip * (16 * NT);
    const int mBase = mPair * 32;
    const int aRow0 = mBase + row16;
    // tail pair (mPair==62): clamp second subtile rows in-bounds, discard its stores
    int aRow1 = mBase + 16 + row16;
    if (aRow1 >= GN) aRow1 = GN - 1;

    v8f zero = {};
    v8f acc0[NT], acc1[NT];
#pragma unroll
    for (int j = 0; j < NT; ++j) { acc0[j] = zero; acc1[j] = zero; }

    const float* Ar0 = A + (size_t)aRow0 * GN + 2 * half;   // K pairs split across lane halves
    const float* Ar1 = A + (size_t)aRow1 * GN + 2 * half;
    for (int k = 0; k < GN; k += 4) {
        v2f a0 = *(const v2f*)(Ar0 + k);
        v2f a1 = *(const v2f*)(Ar1 + k);
        const float* Xp = X + (size_t)(k + 2 * half) * GF + nBase + row16;
#pragma unroll
        for (int j = 0; j < NT; ++j) {
            v2f bfrag;
            bfrag.x = Xp[j * 16];          // K row k+2h
            bfrag.y = Xp[GF + j * 16];     // K row k+2h+1
            acc0[j] = __builtin_amdgcn_wmma_f32_16x16x4_f32(
                false, a0, false, bfrag, (short)0, acc0[j], false, false);
            acc1[j] = __builtin_amdgcn_wmma_f32_16x16x4_f32(
                false, a1, false, bfrag, (short)0, acc1[j], false, false);
        }
    }
    // epilogue: Euler blend, C/D layout: VGPR v -> M = v + 8*half, lanes 0..15 -> N cols
#pragma unroll
    for (int j = 0; j < NT; ++j) {
        const int col = nBase + j * 16 + row16;
#pragma unroll
        for (int v = 0; v < 8; ++v) {
            const int m0 = mBase + v + 8 * half;
            const size_t idx0 = (size_t)m0 * GF + col;
            Y[idx0] = 0.75f * X[idx0] + 0.25f * acc0[j][v];
            const int m1 = mBase + 16 + v + 8 * half;
            if (m1 < GN) {
                const size_t idx1 = (size_t)m1 * GF + col;
                Y[idx1] = 0.75f * X[idx1] + 0.25f * acc1[j][v];
            }
        }
    }
}

// ---------------- 32x32 conv over channels ----------------
// mode 0: G[site,c]  = W @ T + bias
// mode 1: H[site,c] += 0.5*(G[site,c] + W @ T + bias)
__global__ __launch_bounds__(256) void conv_g_kernel(const float* __restrict__ T,
                                                     const float* __restrict__ W,   // [32,32]
                                                     const float* __restrict__ bias,
                                                     const float* __restrict__ G,
                                                     float* __restrict__ Gout,
                                                     float* __restrict__ H,
                                                     int mode) {
    __shared__ float sw[GC * GC], sb[GC];
    for (int i = threadIdx.x; i < GC * GC; i += 256) sw[i] = W[i];
    if (threadIdx.x < GC) sb[threadIdx.x] = bias[threadIdx.x];
    __syncthreads();
    const int sid = blockIdx.x * 256 + threadIdx.x;
    if (sid >= GSITES) return;
    const int n = sid / (GB * GL);
    const int r = sid % (GB * GL);
    const int b = r / GL;
    const int l = r % GL;
    const size_t base = (size_t)n * GF + b * (GC * GL) + l;

    float in[GC];
#pragma unroll
    for (int c = 0; c < GC; ++c) in[c] = T[base + c * GL];

#pragma unroll
    for (int o = 0; o < GC; ++o) {
        float acc = sb[o];
#pragma unroll
        for (int c = 0; c < GC; ++c) acc = fmaf(sw[o * GC + c], in[c], acc);
        if (mode == 0) {
            Gout[base + o * GL] = acc;
        } else {
            H[base + o * GL] += 0.5f * (G[base + o * GL] + acc);
        }
    }
}

// ---------------- layernorm stats per batch ----------------
__global__ __launch_bounds__(256) void ln_stats_kernel(const float* __restrict__ H,
                                                       float* __restrict__ mu,
                                                       float* __restrict__ var) {
    __shared__ double s1[256], s2[256];
    const int b = blockIdx.x;
    double a1 = 0.0, a2 = 0.0;
    for (int n = 0; n < GN; ++n) {
        const float* p = H + (size_t)n * GF + b * (GC * GL);
        for (int j = threadIdx.x; j < GC * GL; j += 256) {
            double v = (double)p[j];
            a1 += v; a2 += v * v;
        }
    }
    s1[threadIdx.x] = a1; s2[threadIdx.x] = a2;
    __syncthreads();
    for (int o = 128; o > 0; o >>= 1) {
        if (threadIdx.x < o) { s1[threadIdx.x] += s1[threadIdx.x + o];
                               s2[threadIdx.x] += s2[threadIdx.x + o]; }
        __syncthreads();
    }
    if (threadIdx.x == 0) {
        const double cnt = (double)GC * GN * GL;
        double m = s1[0] / cnt;
        mu[b]  = (float)m;
        var[b] = (float)(s2[0] / cnt - m * m);
    }
}

// ---------------- fused MLP head: norm -> 32->64 relu -> 64->128 relu -> 128->12 ----------------
__global__ __launch_bounds__(256) void mlp_out_kernel(const float* __restrict__ H,
                                                      const float* __restrict__ mu,
                                                      const float* __restrict__ var,
                                                      const float* __restrict__ we0, const float* __restrict__ be0,
                                                      const float* __restrict__ we1, const float* __restrict__ be1,
                                                      const float* __restrict__ we2, const float* __restrict__ be2,
                                                      float* __restrict__ out) {
    __shared__ float s_e0[64 * 32], s_b0[64];
    __shared__ float s_e1[128 * 64], s_b1[128];
    __shared__ float s_e2[12 * 128], s_b2[12];
    for (int i = threadIdx.x; i < 64 * 32; i += 256) s_e0[i] = we0[i];
    for (int i = threadIdx.x; i < 128 * 64; i += 256) s_e1[i] = we1[i];
    for (int i = threadIdx.x; i < 12 * 128; i += 256) s_e2[i] = we2[i];
    if (threadIdx.x < 64)  s_b0[threadIdx.x] = be0[threadIdx.x];
    if (threadIdx.x < 128) s_b1[threadIdx.x] = be1[threadIdx.x];
    if (threadIdx.x < 12)  s_b2[threadIdx.x] = be2[threadIdx.x];
    __syncthreads();

    const int sid = blockIdx.x * 256 + threadIdx.x;
    if (sid >= GSITES) return;
    const int n = sid / (GB * GL);
    const int r = sid % (GB * GL);
    const int b = r / GL;
    const int l = r % GL;
    const float m  = mu[b];
    const float rs = rsqrtf(var[b] + 1e-5f);

    float h[GC];
    {
        const float* p = H + (size_t)n * GF + b * (GC * GL) + l;
#pragma unroll
        for (int c = 0; c < GC; ++c) h[c] = (p[c * GL] - m) * rs;
    }

    float v0[64];
#pragma unroll
    for (int o = 0; o < 64; ++o) {
        float acc = s_b0[o];
#pragma unroll
        for (int c = 0; c < GC; ++c) acc = fmaf(s_e0[o * GC + c], h[c], acc);
        v0[o] = fmaxf(acc, 0.f);
    }

    float res[GCO];
#pragma unroll
    for (int j = 0; j < GCO; ++j) res[j] = s_b2[j];

    for (int ch = 0; ch < 8; ++ch) {          // 128 v1 entries in chunks of 16
        float t[16];
#pragma unroll
        for (int i = 0; i < 16; ++i) {
            const int o = ch * 16 + i;
            float acc = s_b1[o];
#pragma unroll
            for (int k = 0; k < 64; ++k) acc = fmaf(s_e1[o * 64 + k], v0[k], acc);
            t[i] = fmaxf(acc, 0.f);
        }
#pragma unroll
        for (int j = 0; j < GCO; ++j) {
            float acc = res[j];
#pragma unroll
            for (int i = 0; i < 16; ++i) acc = fmaf(s_e2[j * GE + ch * 16 + i], t[i], acc);
            res[j] = acc;
        }
    }

    float* op = out + ((size_t)((b * GL + l) * GN + n)) * GCO;
#pragma unroll
    for (int j = 0; j < GCO; ++j) op[j] = res[j];
}

// ---------------- host orchestration ----------------
extern "C" void kernel_launch(void* const* d_in, const int* in_sizes, int n_in,
                              void* d_out, int out_size, void* d_ws, size_t ws_size,
                              hipStream_t stream) {
    const float* x       = (const float*)d_in[0];
    const float* adj     = (const float*)d_in[1];
    const float* w_start = (const float*)d_in[2];
    const float* b_start = (const float*)d_in[3];
    const float* w_g1    = (const float*)d_in[4];
    const float* b_g1    = (const float*)d_in[5];
    const float* w_g2    = (const float*)d_in[6];
    const float* b_g2    = (const float*)d_in[7];
    const float* w_e0    = (const float*)d_in[8];
    const float* b_e0    = (const float*)d_in[9];
    const float* w_e1    = (const float*)d_in[10];
    const float* b_e1    = (const float*)d_in[11];
    const float* w_e2    = (const float*)d_in[12];
    const float* b_e2    = (const float*)d_in[13];
    float* out = (float*)d_out;

    float* ws = (float*)d_ws;
    const size_t SZ_A = (size_t)GN * GN;   // 4e6
    const size_t SZ_H = (size_t)GN * GF;   // 12.288e6
    float* A1   = ws;                 ws += SZ_A;
    float* A2   = ws;                 ws += SZ_A;
    float* H    = ws;                 ws += SZ_H;
    float* T0   = ws;                 ws += SZ_H;
    float* T1   = ws;                 ws += SZ_H;
    float* G    = ws;                 ws += SZ_H;
    float* rsum = ws;                 ws += GN;
    float* csum = ws;                 ws += GN;
    float* mu   = ws;                 ws += GB;
    float* var  = ws;                 ws += GB;

    const dim3 blk(256);

    rowsum_kernel<<<GN, blk, 0, stream>>>(adj, rsum);
    colsum_kernel<<<(GN + 255) / 256, blk, 0, stream>>>(adj, csum);
    build_adj_kernel<<<(unsigned)((SZ_A + 255) / 256), blk, 0, stream>>>(adj, rsum, csum, A1, A2);
    conv_start_kernel<<<(GSITES + 255) / 256, blk, 0, stream>>>(x, w_start, b_start, H);

    const dim3 pgrid((GN + 31) / 32, (GF / (16 * NT)) / 8);   // 63 x 6, 8 waves/block
    const int cgrid = (GSITES + 255) / 256;

    for (int t = 0; t < 2; ++t) {
        // cgp with A1
        prop_step_kernel<<<pgrid, blk, 0, stream>>>(A1, H,  T0);
        prop_step_kernel<<<pgrid, blk, 0, stream>>>(A1, T0, T1);
        prop_step_kernel<<<pgrid, blk, 0, stream>>>(A1, T1, T0);
        prop_step_kernel<<<pgrid, blk, 0, stream>>>(A1, T0, T1);
        conv_g_kernel<<<cgrid, blk, 0, stream>>>(T1, w_g1, b_g1, nullptr, G, nullptr, 0);
        // cgp with A2
        prop_step_kernel<<<pgrid, blk, 0, stream>>>(A2, H,  T0);
        prop_step_kernel<<<pgrid, blk, 0, stream>>>(A2, T0, T1);
        prop_step_kernel<<<pgrid, blk, 0, stream>>>(A2, T1, T0);
        prop_step_kernel<<<pgrid, blk, 0, stream>>>(A2, T0, T1);
        // fused outer-Euler update: H += 0.5*(G + conv_g2(T1))
        conv_g_kernel<<<cgrid, blk, 0, stream>>>(T1, w_g2, b_g2, G, nullptr, H, 1);
    }

    ln_stats_kernel<<<GB, blk, 0, stream>>>(H, mu, var);
    mlp_out_kernel<<<(GSITES + 255) / 256, blk, 0, stream>>>(
        H, mu, var, w_e0, b_e0, w_e1, b_e1, w_e2, b_e2, out);
}